// MultiOmicGATModule_17282948399618
// MI455X (gfx1250) — compile-verified
//
#include <hip/hip_runtime.h>
#include <hip/hip_bf16.h>
#include <math.h>

// ---------------------------------------------------------------------------
// MultiOmic hetero-GATv2 for MI455X (gfx1250, wave32).
// Dense node transforms + readout GEMMs run on V_WMMA_F32_16X16X4_F32 (full
// f32 precision; the 52 GFLOP of dense work is trivial next to the L2-resident
// edge gather/scatter traffic, so f32 WMMA is the right precision choice).
// ---------------------------------------------------------------------------

#define HID   256
#define NHEAD 8
#define CHAN  32
#define NGENE 20000
#define NCPG  50000
#define NMIR  2000
#define BATCH 64

typedef float v2f __attribute__((ext_vector_type(2)));
typedef float v8f __attribute__((ext_vector_type(8)));

// ---- float atomic max via sign-magnitude int ordering (always compiles) ----
__device__ __forceinline__ void atomicMaxF(float* addr, float v) {
  if (v >= 0.0f) atomicMax((int*)addr, __float_as_int(v));
  else           atomicMin((unsigned int*)addr, __float_as_uint(v));
}

__device__ __forceinline__ float waveRedSum(float v) {
#pragma unroll
  for (int m = 16; m > 0; m >>= 1) v += __shfl_xor(v, m, 32);
  return v;
}

// ---------------------------- utility kernels ------------------------------
__global__ __launch_bounds__(256) void fill_f32(float* p, float v, int n) {
  int t = blockIdx.x * blockDim.x + threadIdx.x;
  if (t < n) p[t] = v;
}

__global__ __launch_bounds__(256) void copy_f32(float* dst, const float* src, int n) {
  int t = blockIdx.x * blockDim.x + threadIdx.x;
  if (t < n) dst[t] = src[t];
}

// O[i*256+j] = bias_cg[j] + bias_mg[j] + bias_gg[j]  (HeteroConv 'sum' seed)
__global__ __launch_bounds__(256) void init_bias3(float* O, const float* b0,
                                                  const float* b1, const float* b2,
                                                  int n) {
  int t = blockIdx.x * blockDim.x + threadIdx.x;
  if (t < n) { int j = t & (HID - 1); O[t] = b0[j] + b1[j] + b2[j]; }
}

// ------------------- WMMA f32 GEMM: Y[rows,256] = X@W + b ------------------
// One wave computes a 16x16 tile; block = 4 waves = 16 rows x 64 cols.
// grid = (rows/16, 4). A/B operand layouts follow ISA 7.12.2 for 16x16x4 f32:
//   A: lanes 0-15 hold M=row, {K=k,k+1}; lanes 16-31 hold {K=k+2,k+3}
//   B: lanes 0-15 hold N=col, {K=k,k+1}; lanes 16-31 hold {K=k+2,k+3}
__global__ __launch_bounds__(128) void gemm256_bias_wmma(
    const float* __restrict__ X, const float* __restrict__ W,
    const float* __restrict__ bias, float* __restrict__ Y) {
  const int wave = threadIdx.x >> 5;
  const int lane = threadIdx.x & 31;
  const int m0   = blockIdx.x << 4;
  const int n0   = (blockIdx.y << 6) + (wave << 4);
  const int row  = m0 + (lane & 15);
  const int kofs = (lane >> 4) << 1;     // 0 or 2
  const int col  = n0 + (lane & 15);

  const float* xp = X + (size_t)row * HID + kofs;
  const float* wp = W + (size_t)kofs * HID + col;

  v8f acc = {};
#pragma unroll 4
  for (int k = 0; k < HID; k += 4) {
    v2f a = *(const v2f*)(xp + k);                       // 8B-aligned
    v2f b = { wp[(size_t)k * HID], wp[(size_t)k * HID + HID] };
    acc = __builtin_amdgcn_wmma_f32_16x16x4_f32(
        false, a, false, b, (short)0, acc, false, false);
  }
  const float bj = bias[col];
  const int mb = m0 + ((lane >> 4) << 3);
#pragma unroll
  for (int j = 0; j < 8; ++j)
    Y[(size_t)(mb + j) * HID + col] = acc[j] + bj;
}

// ---------------- split-K WMMA readout: Zacc += X[64,N] @ F[N,256] ---------
// grid = (KSPLIT, 4, 4); block = 128 (4 waves, 16 cols each).
__global__ __launch_bounds__(128) void readout_wmma(
    const float* __restrict__ X, const float* __restrict__ F,
    float* __restrict__ Zacc, int Nnodes, int kchunk) {
  const int wave = threadIdx.x >> 5;
  const int lane = threadIdx.x & 31;
  const int m0   = blockIdx.z << 4;
  const int n0   = (blockIdx.y << 6) + (wave << 4);
  const int k0   = blockIdx.x * kchunk;
  const int k1   = (k0 + kchunk < Nnodes) ? (k0 + kchunk) : Nnodes;
  const int row  = m0 + (lane & 15);
  const int kofs = (lane >> 4) << 1;
  const int col  = n0 + (lane & 15);

  v8f acc = {};
  for (int k = k0; k < k1; k += 4) {
    v2f a = *(const v2f*)(X + (size_t)row * Nnodes + k + kofs);
    v2f b = { F[(size_t)(k + kofs) * HID + col],
              F[(size_t)(k + kofs + 1) * HID + col] };
    acc = __builtin_amdgcn_wmma_f32_16x16x4_f32(
        false, a, false, b, (short)0, acc, false, false);
  }
  const int mb = m0 + ((lane >> 4) << 3);
#pragma unroll
  for (int j = 0; j < 8; ++j)
    atomicAdd(&Zacc[(size_t)(mb + j) * HID + col], acc[j]);
}

// ----------------------------- edge kernels --------------------------------
// thread = (edge, head). Edge ids >= ebase are implicit self-loops (src=dst).
__global__ __launch_bounds__(256) void edge_logits(
    const float* __restrict__ hs, const float* __restrict__ hd,
    const int* __restrict__ src, const int* __restrict__ dst,
    const float* __restrict__ att, float* __restrict__ elog,
    float* __restrict__ mbuf, int etot, int ebase) {
  int t = blockIdx.x * blockDim.x + threadIdx.x;
  if (t >= etot * NHEAD) return;
  int eid = t >> 3, h = t & 7;
  int s, d;
  if (eid < ebase) { s = src[eid]; d = dst[eid]; } else { s = d = eid - ebase; }
  const float4* pa = (const float4*)(hs + (size_t)s * HID + h * CHAN);
  const float4* pb = (const float4*)(hd + (size_t)d * HID + h * CHAN);
  const float4* pc = (const float4*)(att + h * CHAN);
  float acc = 0.f;
#pragma unroll
  for (int q = 0; q < 8; ++q) {
    float4 a = pa[q], b = pb[q], c = pc[q];
    float x;
    x = a.x + b.x; acc += (x > 0.f ? x : 0.2f * x) * c.x;
    x = a.y + b.y; acc += (x > 0.f ? x : 0.2f * x) * c.y;
    x = a.z + b.z; acc += (x > 0.f ? x : 0.2f * x) * c.z;
    x = a.w + b.w; acc += (x > 0.f ? x : 0.2f * x) * c.w;
  }
  elog[t] = acc;
  atomicMaxF(mbuf + (size_t)d * NHEAD + h, acc);
}

__global__ __launch_bounds__(256) void edge_expsum(
    float* __restrict__ elog, const int* __restrict__ dst,
    const float* __restrict__ mbuf, float* __restrict__ den,
    int etot, int ebase) {
  int t = blockIdx.x * blockDim.x + threadIdx.x;
  if (t >= etot * NHEAD) return;
  int eid = t >> 3, h = t & 7;
  int d = (eid < ebase) ? dst[eid] : (eid - ebase);
  float ex = expf(elog[t] - mbuf[(size_t)d * NHEAD + h]);
  elog[t] = ex;
  atomicAdd(den + (size_t)d * NHEAD + h, ex);
}

// block = one edge (256 threads = 256 channels); scatter alpha*hs[src] into O.
__global__ __launch_bounds__(256) void edge_scatter(
    const float* __restrict__ ex, const float* __restrict__ den,
    const float* __restrict__ hs,
    const int* __restrict__ src, const int* __restrict__ dst,
    float* __restrict__ O, int etot, int ebase) {
  int eid = blockIdx.x;
  if (eid >= etot) return;
  int j = threadIdx.x;
  int h = j >> 5;
  int s, d;
  if (eid < ebase) { s = src[eid]; d = dst[eid]; } else { s = d = eid - ebase; }
  float alpha = ex[(size_t)eid * NHEAD + h] / (den[(size_t)d * NHEAD + h] + 1e-16f);
  atomicAdd(O + (size_t)d * HID + j, alpha * hs[(size_t)s * HID + j]);
}

// -------------------- g_next = LN(elu(O) + g) , wave per node --------------
__global__ __launch_bounds__(256) void elu_res_ln(
    const float* __restrict__ O, const float* __restrict__ Gin,
    const float* __restrict__ gamma, const float* __restrict__ beta,
    float* __restrict__ Gout, int n) {
  int wave = threadIdx.x >> 5, lane = threadIdx.x & 31;
  int node = blockIdx.x * 8 + wave;
  if (node >= n) return;
  const float* op = O   + (size_t)node * HID;
  const float* gp = Gin + (size_t)node * HID;
  float v[8];
  float s = 0.f;
#pragma unroll
  for (int q = 0; q < 8; ++q) {
    float o = op[lane + 32 * q];
    o = (o > 0.f) ? o : (expf(o) - 1.f);   // ELU(alpha=1)
    v[q] = o + gp[lane + 32 * q];
    s += v[q];
  }
  float mu = waveRedSum(s) * (1.f / 256.f);
  float vs = 0.f;
#pragma unroll
  for (int q = 0; q < 8; ++q) { float dlt = v[q] - mu; vs += dlt * dlt; }
  float var = waveRedSum(vs) * (1.f / 256.f);
  float r = rsqrtf(var + 1e-5f);
#pragma unroll
  for (int q = 0; q < 8; ++q) {
    int j = lane + 32 * q;
    Gout[(size_t)node * HID + j] = (v[q] - mu) * r * gamma[j] + beta[j];
  }
}

// ------------- out = LN(Z*scale) per row (64 rows), wave per row -----------
__global__ __launch_bounds__(256) void scale_ln_rows(
    const float* __restrict__ Z, const float* __restrict__ gamma,
    const float* __restrict__ beta, float* __restrict__ out,
    float scale, int rows) {
  int wave = threadIdx.x >> 5, lane = threadIdx.x & 31;
  int r0 = blockIdx.x * 8 + wave;
  if (r0 >= rows) return;
  float v[8];
  float s = 0.f;
#pragma unroll
  for (int q = 0; q < 8; ++q) {
    v[q] = Z[(size_t)r0 * HID + lane + 32 * q] * scale;
    s += v[q];
  }
  float mu = waveRedSum(s) * (1.f / 256.f);
  float vs = 0.f;
#pragma unroll
  for (int q = 0; q < 8; ++q) { float dlt = v[q] - mu; vs += dlt * dlt; }
  float var = waveRedSum(vs) * (1.f / 256.f);
  float rr = rsqrtf(var + 1e-5f);
#pragma unroll
  for (int q = 0; q < 8; ++q) {
    int j = lane + 32 * q;
    out[(size_t)r0 * HID + j] = (v[q] - mu) * rr * gamma[j] + beta[j];
  }
}

// ---------------------------------------------------------------------------
extern "C" void kernel_launch(void* const* d_in, const int* in_sizes, int n_in,
                              void* d_out, int out_size, void* d_ws, size_t ws_size,
                              hipStream_t stream) {
  (void)n_in; (void)out_size; (void)ws_size;
  // Leaf order assumption: setup_inputs() insertion order, params flattened
  // recursively in insertion order:
  //  0:xg 1:xc 2:xm 3:cg_src 4:cg_dst 5:mg_src 6:mg_dst 7:gg_src 8:gg_dst
  //  9:emb_gene 10:emb_cpg 11:emb_mir
  //  per layer L (base=12+20L): cg{Wl,bl,Wr,br,att,bias} mg{..+6} gg{..+12}
  //                             ln_g(+18) ln_b(+19)
  //  72..77: on_gene(g,b) on_cpg(g,b) on_mir(g,b)
  const float* xg = (const float*)d_in[0];
  const float* xc = (const float*)d_in[1];
  const float* xm = (const float*)d_in[2];
  const int* cg_src = (const int*)d_in[3];
  const int* cg_dst = (const int*)d_in[4];
  const int* mg_src = (const int*)d_in[5];
  const int* mg_dst = (const int*)d_in[6];
  const int* gg_src = (const int*)d_in[7];
  const int* gg_dst = (const int*)d_in[8];
  const int E_cg = in_sizes[3], E_mg = in_sizes[5], E_gg = in_sizes[7];
  auto F = [&](int i) { return (const float*)d_in[i]; };
  const float* emb_gene = F(9);
  const float* emb_cpg  = F(10);
  const float* emb_mir  = F(11);

  // ---- workspace layout (floats) ----
  float* ws = (float*)d_ws;
  size_t off = 0;
  float* G0   = ws + off; off += (size_t)NGENE * HID;
  float* G1   = ws + off; off += (size_t)NGENE * HID;
  float* HS   = ws + off; off += (size_t)NCPG * HID;      // max source count
  float* HD   = ws + off; off += (size_t)NGENE * HID;
  float* OACC = ws + off; off += (size_t)NGENE * HID;
  int emax = E_cg; if (E_mg > emax) emax = E_mg;
  if (E_gg + NGENE > emax) emax = E_gg + NGENE;
  float* ELOG = ws + off; off += (size_t)emax * NHEAD;
  float* MBUF = ws + off; off += (size_t)NGENE * NHEAD;
  float* DEN  = ws + off; off += (size_t)NGENE * NHEAD;
  float* ZACC = ws + off; off += (size_t)3 * BATCH * HID;

  auto gemm = [&](const float* X, const float* W, const float* bv, float* Y, int rows) {
    gemm256_bias_wmma<<<dim3(rows / 16, 4, 1), 128, 0, stream>>>(X, W, bv, Y);
  };

  auto relation = [&](const float* src_feat, int nsrc, const float* g_cur,
                      const int* esrc, const int* edst, int E, int nself,
                      const float* Wl, const float* bl,
                      const float* Wr, const float* br, const float* att) {
    gemm(src_feat, Wl, bl, HS, nsrc);
    gemm(g_cur, Wr, br, HD, NGENE);
    int etot = E + nself;
    int nmh = NGENE * NHEAD;
    fill_f32<<<(nmh + 255) / 256, 256, 0, stream>>>(MBUF, -3.0e38f, nmh);
    fill_f32<<<(nmh + 255) / 256, 256, 0, stream>>>(DEN, 0.f, nmh);
    int nth = etot * NHEAD;
    edge_logits<<<(nth + 255) / 256, 256, 0, stream>>>(HS, HD, esrc, edst, att,
                                                       ELOG, MBUF, etot, E);
    edge_expsum<<<(nth + 255) / 256, 256, 0, stream>>>(ELOG, edst, MBUF, DEN,
                                                       etot, E);
    edge_scatter<<<etot, 256, 0, stream>>>(ELOG, DEN, HS, esrc, edst, OACC,
                                           etot, E);
  };

  // ---- g = emb_gene ----
  copy_f32<<<(NGENE * HID + 255) / 256, 256, 0, stream>>>(G0, emb_gene, NGENE * HID);
  float* gc = G0;
  float* gn = G1;

  for (int L = 0; L < 3; ++L) {
    int base = 12 + L * 20;
    const float *Wl_cg = F(base + 0),  *bl_cg = F(base + 1),
                *Wr_cg = F(base + 2),  *br_cg = F(base + 3),
                *at_cg = F(base + 4),  *bi_cg = F(base + 5);
    const float *Wl_mg = F(base + 6),  *bl_mg = F(base + 7),
                *Wr_mg = F(base + 8),  *br_mg = F(base + 9),
                *at_mg = F(base + 10), *bi_mg = F(base + 11);
    const float *Wl_gg = F(base + 12), *bl_gg = F(base + 13),
                *Wr_gg = F(base + 14), *br_gg = F(base + 15),
                *at_gg = F(base + 16), *bi_gg = F(base + 17);
    const float *ln_g = F(base + 18), *ln_b = F(base + 19);

    init_bias3<<<(NGENE * HID + 255) / 256, 256, 0, stream>>>(OACC, bi_cg, bi_mg,
                                                              bi_gg, NGENE * HID);
    relation(emb_cpg, NCPG, gc, cg_src, cg_dst, E_cg, 0,
             Wl_cg, bl_cg, Wr_cg, br_cg, at_cg);
    relation(emb_mir, NMIR, gc, mg_src, mg_dst, E_mg, 0,
             Wl_mg, bl_mg, Wr_mg, br_mg, at_mg);
    relation(gc, NGENE, gc, gg_src, gg_dst, E_gg, NGENE,
             Wl_gg, bl_gg, Wr_gg, br_gg, at_gg);

    elu_res_ln<<<NGENE / 8, 256, 0, stream>>>(OACC, gc, ln_g, ln_b, gn, NGENE);
    float* t = gc; gc = gn; gn = t;
  }

  // ---- readout GEMMs (split-K=25) + LayerNorm ----
  float* out = (float*)d_out;
  fill_f32<<<(3 * BATCH * HID + 255) / 256, 256, 0, stream>>>(ZACC, 0.f,
                                                              3 * BATCH * HID);
  readout_wmma<<<dim3(25, 4, 4), 128, 0, stream>>>(xg, gc, ZACC, NGENE, NGENE / 25);
  readout_wmma<<<dim3(25, 4, 4), 128, 0, stream>>>(xc, emb_cpg, ZACC + BATCH * HID,
                                                   NCPG, NCPG / 25);
  readout_wmma<<<dim3(25, 4, 4), 128, 0, stream>>>(xm, emb_mir, ZACC + 2 * BATCH * HID,
                                                   NMIR, NMIR / 25);
  scale_ln_rows<<<8, 256, 0, stream>>>(ZACC, F(72), F(73), out,
                                       1.f / sqrtf((float)NGENE), BATCH);
  scale_ln_rows<<<8, 256, 0, stream>>>(ZACC + BATCH * HID, F(74), F(75),
                                       out + BATCH * HID,
                                       1.f / sqrtf((float)NCPG), BATCH);
  scale_ln_rows<<<8, 256, 0, stream>>>(ZACC + 2 * BATCH * HID, F(76), F(77),
                                       out + 2 * BATCH * HID,
                                       1.f / sqrtf((float)NMIR), BATCH);
}